// CapsuleLikelihoodTorch_19619410608286
// MI455X (gfx1250) — compile-verified
//
#include <hip/hip_runtime.h>
#include <math.h>

typedef __attribute__((ext_vector_type(2))) float v2f;
typedef __attribute__((ext_vector_type(8))) float v8f;

#define CV 512            // C*V = 32*16 Gaussians per example
#define TILE_PTS 16
#define WAVES 4
#define JT_PER_WAVE 8     // 512 / (16 cols * 4 waves)
#define MS_STRIDE 9       // pad to 9 floats: stride 9 vs 64 banks -> conflict free
#define LOG_2PI 1.8378770664093453f

// DPP16 controls for a 16-lane butterfly reduction (VALU-only, no LDS pipe):
//   0xB1  = quad_perm [1,0,3,2]  (xor 1)
//   0x4E  = quad_perm [2,3,0,1]  (xor 2)
//   0x141 = row_half_mirror      (combines 4-apart after quads)
//   0x140 = row_mirror           (combines 8-apart; stays within 16-lane row)
#define DPP_F32(v, ctrl)                                                        \
  __builtin_bit_cast(float, __builtin_amdgcn_mov_dpp(                           \
      __builtin_bit_cast(int, (v)), (ctrl), 0xf, 0xf, true))

__device__ __forceinline__ float dpp_red_max16(float v) {
  v = fmaxf(v, DPP_F32(v, 0xB1));
  v = fmaxf(v, DPP_F32(v, 0x4E));
  v = fmaxf(v, DPP_F32(v, 0x141));
  v = fmaxf(v, DPP_F32(v, 0x140));
  return v;
}

__device__ __forceinline__ float dpp_red_sum16(float v) {
  v += DPP_F32(v, 0xB1);
  v += DPP_F32(v, 0x4E);
  v += DPP_F32(v, 0x141);
  v += DPP_F32(v, 0x140);
  return v;
}

__global__ void caps_zero_kernel(float* out, int n) {
  int i = blockIdx.x * blockDim.x + threadIdx.x;
  if (i < n) out[i] = 0.0f;
}

__global__ __launch_bounds__(128)
void caps_main_kernel(const float* __restrict__ x,
                      const float* __restrict__ vote,
                      const float* __restrict__ scl,
                      const float* __restrict__ lpr,
                      const int* __restrict__ batch,
                      float* __restrict__ out, int N) {
  __shared__ float Ms[CV * MS_STRIDE];          // transformed Gaussian table (18KB)
  __shared__ float Xs[TILE_PTS * 8];            // extended point tile
  __shared__ float red_m[WAVES][TILE_PTS];
  __shared__ float red_s[WAVES][TILE_PTS];
  __shared__ float lps[TILE_PTS];

  const int tid  = threadIdx.x;
  const int wave = tid >> 5;
  const int lane = tid & 31;
  const int p0   = blockIdx.x * TILE_PTS;

  // Stage extended point rows {x0..x5, |x|^2, 1}
  if (tid < TILE_PTS) {
    int p = min(p0 + tid, N - 1);
    const float* xr = x + (size_t)p * 6;
    float x0=xr[0],x1=xr[1],x2=xr[2],x3=xr[3],x4=xr[4],x5=xr[5];
    float* row = Xs + tid * 8;
    row[0]=x0; row[1]=x1; row[2]=x2; row[3]=x3; row[4]=x4; row[5]=x5;
    row[6]=x0*x0+x1*x1+x2*x2+x3*x3+x4*x4+x5*x5;
    row[7]=1.0f;
  }
  const int b_lo = batch[min(p0, N - 1)];
  const int b_hi = batch[min(p0 + TILE_PTS - 1, N - 1)];
  __syncthreads();

  // wave32 f32 WMMA operand mapping (ISA 7.12.2):
  // A 16x4: lanes 0-15 rows M, VGPR0/1 = K0/K1; lanes 16-31 = K2/K3
  // B 4x16: lanes 0-15 cols N, VGPR0/1 = K0/K1; lanes 16-31 = K2/K3
  const int mrow = lane & 15;
  const int kh   = (lane >> 4) << 1;

  for (int b = b_lo; b <= b_hi; ++b) {
    // Stage transformed Gaussians for example b:
    // {mu*inv_s2 (6), -0.5*inv_s2, log_pres - 6*log(s) - 3*log(2pi) - 0.5*inv_s2*|mu|^2}
    for (int j = tid; j < CV; j += 128) {
      size_t gi = (size_t)b * CV + j;
      const float* mu = vote + gi * 6;
      float m0=mu[0],m1=mu[1],m2=mu[2],m3=mu[3],m4=mu[4],m5=mu[5];
      float s  = fmaxf(scl[gi], 1e-10f);
      float iv = 1.0f / s;
      float i2 = iv * iv;
      float n2 = m0*m0+m1*m1+m2*m2+m3*m3+m4*m4+m5*m5;
      float cst = lpr[gi] - 6.0f * __logf(s) - 3.0f * LOG_2PI - 0.5f * i2 * n2;
      float* row = Ms + j * MS_STRIDE;
      row[0]=m0*i2; row[1]=m1*i2; row[2]=m2*i2;
      row[3]=m3*i2; row[4]=m4*i2; row[5]=m5*i2;
      row[6]=-0.5f * i2; row[7]=cst;
    }
    __syncthreads();

    v2f a0, a1;
    a0.x = Xs[mrow*8 + kh];     a0.y = Xs[mrow*8 + kh + 1];
    a1.x = Xs[mrow*8 + 4 + kh]; a1.y = Xs[mrow*8 + 4 + kh + 1];

    // logit[point_row, col] = dot8(xt, mt) via two chained f32 WMMAs per tile
    float lg[JT_PER_WAVE * 8];
    #pragma unroll
    for (int t = 0; t < JT_PER_WAVE; ++t) {
      int jj = (wave * JT_PER_WAVE + t) * 16 + mrow;
      const float* br = Ms + jj * MS_STRIDE;
      v2f b0, b1;
      b0.x = br[kh];     b0.y = br[kh + 1];
      b1.x = br[4 + kh]; b1.y = br[4 + kh + 1];
      v8f c = {0.f,0.f,0.f,0.f,0.f,0.f,0.f,0.f};
      c = __builtin_amdgcn_wmma_f32_16x16x4_f32(false, a0, false, b0, (short)0, c, false, false);
      c = __builtin_amdgcn_wmma_f32_16x16x4_f32(false, a1, false, b1, (short)0, c, false, false);
      #pragma unroll
      for (int r = 0; r < 8; ++r) lg[t*8 + r] = c[r];
    }

    // Per-point logsumexp: C/D layout puts row M=r(+8*half) in VGPR r, col N=lane&15.
    // Reduce over tiles in-lane, then across the 16-lane group with DPP16 (VALU only).
    #pragma unroll
    for (int r = 0; r < 8; ++r) {
      float m = lg[r];
      #pragma unroll
      for (int t = 1; t < JT_PER_WAVE; ++t) m = fmaxf(m, lg[t*8 + r]);
      m = dpp_red_max16(m);
      float s = 0.0f;
      #pragma unroll
      for (int t = 0; t < JT_PER_WAVE; ++t) s += __expf(lg[t*8 + r] - m);
      s = dpp_red_sum16(s);
      if (mrow == 0) {
        int prow = r + ((lane >> 4) << 3);
        red_m[wave][prow] = m;
        red_s[wave][prow] = s;
      }
    }
    __syncthreads();

    // Combine 4 wave partials per point; mask padded / other-batch points
    if (tid < TILE_PTS) {
      float M = red_m[0][tid];
      for (int w = 1; w < WAVES; ++w) M = fmaxf(M, red_m[w][tid]);
      float S = 0.0f;
      for (int w = 0; w < WAVES; ++w) S += red_s[w][tid] * __expf(red_m[w][tid] - M);
      float lp = M + __logf(S);
      int p = p0 + tid;
      bool valid = (p < N) && (batch[p] == b);
      lps[tid] = valid ? lp : 0.0f;
    }
    __syncthreads();
    if (tid == 0) {
      float sum = 0.0f;
      #pragma unroll
      for (int i = 0; i < TILE_PTS; ++i) sum += lps[i];
      atomicAdd(out + 1 + b, sum);
    }
    __syncthreads();
  }
}

__global__ void caps_final_kernel(float* out, int B) {
  if (threadIdx.x == 0) {
    float s = 0.0f;
    for (int i = 0; i < B; ++i) s += out[1 + i];
    out[0] = s / (float)B;
  }
}

extern "C" void kernel_launch(void* const* d_in, const int* in_sizes, int n_in,
                              void* d_out, int out_size, void* d_ws, size_t ws_size,
                              hipStream_t stream) {
  const float* x    = (const float*)d_in[0];
  const float* vote = (const float*)d_in[1];
  const float* scl  = (const float*)d_in[2];
  const float* lpr  = (const float*)d_in[3];
  const int*   bat  = (const int*)d_in[4];
  float* out = (float*)d_out;

  int N = in_sizes[0] / 6;        // 32768
  int B = in_sizes[2] / CV;       // 32

  caps_zero_kernel<<<(out_size + 63) / 64, 64, 0, stream>>>(out, out_size);
  int grid = (N + TILE_PTS - 1) / TILE_PTS;
  caps_main_kernel<<<grid, 128, 0, stream>>>(x, vote, scl, lpr, bat, out, N);
  caps_final_kernel<<<1, 32, 0, stream>>>(out, B);
}